// LocallyConnected2d_78159814853065
// MI455X (gfx1250) — compile-verified
//
#include <hip/hip_runtime.h>

typedef float v2f __attribute__((ext_vector_type(2)));
typedef float v8f __attribute__((ext_vector_type(8)));

#define Bn    32          // batch
#define Cin   32
#define Hh    64
#define Ww    64
#define Cout  64
#define OHh   62
#define OWw   62
#define Pdim  288         // Cin * 3 * 3 (im2col feature dim, order (c,ki,kj))
#define LDSR  292         // padded LDS row stride (floats): 16B-aligned, conflict-free
#define KSTEPS 72         // Pdim / 4 (one V_WMMA_F32_16X16X4_F32 per step)

__global__ __launch_bounds__(256)
void lc2d_wmma_f32_kernel(const float* __restrict__ x,
                          const float* __restrict__ weight,
                          const float* __restrict__ bias,
                          float* __restrict__ out)
{
    // [n][k] weight slab and [b][k] patch slab for this output location
    __shared__ __align__(16) float ldsB[Cout * LDSR];   // 74,752 B
    __shared__ __align__(16) float ldsA[Bn   * LDSR];   // 37,376 B

    const int tid = threadIdx.x;
    const int hw  = blockIdx.x;           // 0 .. OH*OW-1
    const int h   = hw / OWw;
    const int w   = hw - h * OWw;

    // ---- cooperative load: untied weights for this (h,w) -> LDS (float4) ----
    // weight[o][h][w][p] = weight[o*OH*OW*P + hw*P + p]
    {
        const float* wbase = weight + (size_t)hw * Pdim;
        for (int i = tid; i < Cout * (Pdim / 4); i += 256) {
            const int o = i / (Pdim / 4);
            const int p = (i - o * (Pdim / 4)) * 4;
            const float4 v = *(const float4*)(wbase + (size_t)o * (OHh * OWw * Pdim) + p);
            *(float4*)(&ldsB[o * LDSR + p]) = v;
        }
    }

    // ---- cooperative load: im2col patches for this (h,w) -> LDS ----
    // task t over (b, c, ki): 32*32*3 = 3072 tasks, 3 contiguous floats each
    for (int t = tid; t < Bn * Cin * 3; t += 256) {
        const int ki = t % 3;
        const int bc = t / 3;                 // b*Cin + c
        const int b  = bc / Cin;
        const int c  = bc - b * Cin;
        const float* src = x + (size_t)bc * (Hh * Ww) + (h + ki) * Ww + w;
        float* dst = &ldsA[b * LDSR + c * 9 + ki * 3];
        dst[0] = src[0];
        dst[1] = src[1];
        dst[2] = src[2];
    }

    __syncthreads();

    // ---- per-wave 16x16 output tile: D = A(16x288) * B(288x16) + bias ----
    const int wave = tid >> 5;
    const int lane = tid & 31;
    const int l16  = lane & 15;
    const int koff = (lane & 16) ? 2 : 0;     // A/B fragment K offset per ISA layout
    const int mt   = wave & 1;                // batch-dim tile (2 x 16)
    const int nt   = wave >> 1;               // Cout-dim tile (4 x 16)

    const float* pa = &ldsA[(mt * 16 + l16) * LDSR + koff];
    const float* pb = &ldsB[(nt * 16 + l16) * LDSR + koff];

    // bias depends only on the output channel (lane column n)
    const int   n  = nt * 16 + l16;
    const float bv = bias[(size_t)n * (OHh * OWw) + hw];

    v8f c0 = {bv, bv, bv, bv, bv, bv, bv, bv};
    v8f c1 = {};
    v8f c2 = {};
    v8f c3 = {};

    #pragma unroll
    for (int kk = 0; kk < KSTEPS; kk += 4) {
        v2f a0 = *(const v2f*)(pa + 4 * (kk + 0));
        v2f b0 = *(const v2f*)(pb + 4 * (kk + 0));
        c0 = __builtin_amdgcn_wmma_f32_16x16x4_f32(false, a0, false, b0,
                                                   (short)0, c0, false, false);
        v2f a1 = *(const v2f*)(pa + 4 * (kk + 1));
        v2f b1 = *(const v2f*)(pb + 4 * (kk + 1));
        c1 = __builtin_amdgcn_wmma_f32_16x16x4_f32(false, a1, false, b1,
                                                   (short)0, c1, false, false);
        v2f a2 = *(const v2f*)(pa + 4 * (kk + 2));
        v2f b2 = *(const v2f*)(pb + 4 * (kk + 2));
        c2 = __builtin_amdgcn_wmma_f32_16x16x4_f32(false, a2, false, b2,
                                                   (short)0, c2, false, false);
        v2f a3 = *(const v2f*)(pa + 4 * (kk + 3));
        v2f b3 = *(const v2f*)(pb + 4 * (kk + 3));
        c3 = __builtin_amdgcn_wmma_f32_16x16x4_f32(false, a3, false, b3,
                                                   (short)0, c3, false, false);
    }
    c0 += c1;
    c2 += c3;
    c0 += c2;

    // ---- store D: lane holds col n; VGPR r -> row m = mt*16 + (lane>=16?8:0) + r
    const int mbase = mt * 16 + ((lane & 16) ? 8 : 0);
    #pragma unroll
    for (int r = 0; r < 8; ++r) {
        const int m = mbase + r;   // batch index
        out[(size_t)(m * Cout + n) * (OHh * OWw) + hw] = c0[r];
    }
}

extern "C" void kernel_launch(void* const* d_in, const int* in_sizes, int n_in,
                              void* d_out, int out_size, void* d_ws, size_t ws_size,
                              hipStream_t stream) {
    (void)in_sizes; (void)n_in; (void)d_ws; (void)ws_size; (void)out_size;
    const float* x      = (const float*)d_in[0];
    const float* weight = (const float*)d_in[1];
    const float* bias   = (const float*)d_in[2];
    float*       out    = (float*)d_out;

    dim3 grid(OHh * OWw);   // one workgroup per output spatial location
    dim3 block(256);        // 8 waves: 2 (batch tiles) x 4 (channel tiles)
    lc2d_wmma_f32_kernel<<<grid, block, 0, stream>>>(x, weight, bias, out);
}